// GraphSAGE_90924457656617
// MI455X (gfx1250) — compile-verified
//
#include <hip/hip_runtime.h>

typedef __attribute__((ext_vector_type(2))) float v2f;
typedef __attribute__((ext_vector_type(8))) float v8f;

#define DFEAT  128   // feature width of x and h (both layers aggregate over 128 cols)
#define LDSPAD 130   // LDS row stride (floats): 128 + 2 pad -> conflict-free ds_load_b64

// ---------------------------------------------------------------------------
// float atomic max via signed/unsigned int trick (correct for all finite
// floats and +/-0 given buffer initialized to -inf bit pattern 0xFF800000).
// ---------------------------------------------------------------------------
__device__ __forceinline__ void atomic_max_float(float* addr, float val) {
  if (__float_as_int(val) >= 0) {
    atomicMax((int*)addr, __float_as_int(val));          // +0 and positives
  } else {
    atomicMin((unsigned int*)addr, __float_as_uint(val)); // negatives and -0
  }
}

// ---------------------------------------------------------------------------
// Fill buffer with -inf bit pattern (scatter-max identity).
// ---------------------------------------------------------------------------
__global__ void fill_neginf_kernel(unsigned int* __restrict__ p, long n) {
  long i = (long)blockIdx.x * blockDim.x + threadIdx.x;
  const long stride = (long)gridDim.x * blockDim.x;
  for (; i < n; i += stride) p[i] = 0xFF800000u;
}

__global__ void zero_kernel(float* __restrict__ p, int n) {
  int i = blockIdx.x * blockDim.x + threadIdx.x;
  if (i < n) p[i] = 0.0f;
}

// ---------------------------------------------------------------------------
// Scatter-max: one wave (32 lanes) per edge, 4 consecutive floats per lane.
// Gather of x[src] is a fully coalesced 512B burst per edge.
// ---------------------------------------------------------------------------
__global__ void scatter_max_kernel(const float* __restrict__ x,
                                   const int* __restrict__ src,
                                   const int* __restrict__ dst,
                                   float* __restrict__ agg, int nedges) {
  int t = blockIdx.x * blockDim.x + threadIdx.x;
  int e = t >> 5;
  if (e >= nedges) return;
  int f0 = (t & 31) * 4;
  int s = src[e];
  int d = dst[e];
  const float4 v = *(const float4*)(x + (long)s * DFEAT + f0);
  float* a = agg + (long)d * DFEAT + f0;
  atomic_max_float(a + 0, v.x);
  atomic_max_float(a + 1, v.y);
  atomic_max_float(a + 2, v.z);
  atomic_max_float(a + 3, v.w);
}

// ---------------------------------------------------------------------------
// Fused SAGE linear layer using V_WMMA_F32_16X16X4_F32 (full f32 fidelity):
//   out[M, NOUT] = where(agg==-inf, 0, agg) @ Wl^T + bias + xin @ Wr^T
// One wave computes a 16-row strip across all NOUT columns.
// Weights staged in LDS (padded rows -> conflict-free ds_load_b64).
// Optionally accumulates per-column sum / sum-of-squares for BatchNorm.
//
// 16x4 f32 A layout: lanes 0-15 row M, VGPR{0,1}=K{0,1}; lanes 16-31 K{2,3}.
// 4x16 f32 B layout mirrors A in K;  C/D: vgpr r, lane L -> (M=r+8*(L>=16), N=L&15).
// ---------------------------------------------------------------------------
template <int NOUT>
__global__ void sage_gemm_wmma_kernel(const float* __restrict__ agg,
                                      const float* __restrict__ xin,
                                      const float* __restrict__ Wl,
                                      const float* __restrict__ Wr,
                                      const float* __restrict__ bias,
                                      float* __restrict__ outp,
                                      float* __restrict__ colsum,
                                      float* __restrict__ colsumsq,
                                      int nrows) {
  extern __shared__ float lds[];
  float* ldsWl = lds;                  // [NOUT][LDSPAD]
  float* ldsWr = lds + NOUT * LDSPAD;  // [NOUT][LDSPAD]

  // Cooperative stage of both weight matrices into LDS.
  for (int i = threadIdx.x; i < NOUT * DFEAT; i += blockDim.x) {
    int n = i >> 7;    // / 128
    int k = i & 127;
    ldsWl[n * LDSPAD + k] = Wl[i];
    ldsWr[n * LDSPAD + k] = Wr[i];
  }
  __syncthreads();

  const int wave   = threadIdx.x >> 5;
  const int lane   = threadIdx.x & 31;
  const int tile   = blockIdx.x * (blockDim.x >> 5) + wave;
  const int ntiles = nrows >> 4;
  if (tile >= ntiles) return;          // wave-uniform: EXEC stays all-ones for WMMA

  const int row0 = tile * 16;
  const int m    = lane & 15;          // A row / B column within tile
  const int koff = (lane >> 4) * 2;    // lanes 16-31 carry K+2,K+3

  v8f acc[NOUT / 16] = {};

  const float* Arow0 = agg + (long)(row0 + m) * DFEAT;
  const float* Arow1 = xin + (long)(row0 + m) * DFEAT;

#pragma unroll 2
  for (int ks = 0; ks < DFEAT / 4; ++ks) {
    const int k0 = ks * 4 + koff;
    v2f a0 = *(const v2f*)(Arow0 + k0);
    // isolated nodes: -inf sentinel -> 0 (PyG convention)
    a0.x = (a0.x == -__builtin_inff()) ? 0.0f : a0.x;
    a0.y = (a0.y == -__builtin_inff()) ? 0.0f : a0.y;
    v2f a1 = *(const v2f*)(Arow1 + k0);
#pragma unroll
    for (int t = 0; t < NOUT / 16; ++t) {
      const int n = t * 16 + m;
      v2f b0 = *(const v2f*)(ldsWl + n * LDSPAD + k0);
      v2f b1 = *(const v2f*)(ldsWr + n * LDSPAD + k0);
      acc[t] = __builtin_amdgcn_wmma_f32_16x16x4_f32(
          false, a0, false, b0, (short)0, acc[t], false, false);
      acc[t] = __builtin_amdgcn_wmma_f32_16x16x4_f32(
          false, a1, false, b1, (short)0, acc[t], false, false);
    }
  }

  // Epilogue: add bias, store, and accumulate BN column statistics.
  const int mbase = row0 + (lane >> 4) * 8;
#pragma unroll
  for (int t = 0; t < NOUT / 16; ++t) {
    const int n  = t * 16 + m;
    const float bn = bias[n];
    float s = 0.0f, sq = 0.0f;
#pragma unroll
    for (int r = 0; r < 8; ++r) {
      float v = acc[t][r] + bn;
      outp[(long)(mbase + r) * NOUT + n] = v;
      s  += v;
      sq += v * v;
    }
    if (colsum != nullptr) {
      // fold the two half-wave row groups (same column n in lanes L and L+16)
      s  += __shfl_xor(s, 16, 32);
      sq += __shfl_xor(sq, 16, 32);
      if (lane < 16) {
        atomicAdd(&colsum[n], s);
        atomicAdd(&colsumsq[n], sq);
      }
    }
  }
}

// ---------------------------------------------------------------------------
// BN: turn (sum, sumsq) into per-column scale/shift.
// ---------------------------------------------------------------------------
__global__ void bn_finalize_kernel(const float* __restrict__ colsum,
                                   const float* __restrict__ colsumsq,
                                   const float* __restrict__ gamma,
                                   const float* __restrict__ beta,
                                   float* __restrict__ scale,
                                   float* __restrict__ shift, int nrows) {
  int n = threadIdx.x;
  if (n < DFEAT) {
    float inv  = 1.0f / (float)nrows;
    float mean = colsum[n] * inv;
    float var  = colsumsq[n] * inv - mean * mean;
    float sc   = gamma[n] * rsqrtf(var + 1e-5f);
    scale[n] = sc;
    shift[n] = beta[n] - mean * sc;
  }
}

// ---------------------------------------------------------------------------
// In-place BN affine + ReLU over h_pre -> h. float4 per thread.
// ---------------------------------------------------------------------------
__global__ void bn_relu_kernel(float* __restrict__ h,
                               const float* __restrict__ scale,
                               const float* __restrict__ shift, long nelem) {
  long i = ((long)blockIdx.x * blockDim.x + threadIdx.x) * 4;
  if (i >= nelem) return;
  int c = (int)(i & (DFEAT - 1));
  float4 v = *(float4*)(h + i);
  v.x = fmaxf(v.x * scale[c + 0] + shift[c + 0], 0.0f);
  v.y = fmaxf(v.y * scale[c + 1] + shift[c + 1], 0.0f);
  v.z = fmaxf(v.z * scale[c + 2] + shift[c + 2], 0.0f);
  v.w = fmaxf(v.w * scale[c + 3] + shift[c + 3], 0.0f);
  *(float4*)(h + i) = v;
}

// ---------------------------------------------------------------------------
extern "C" void kernel_launch(void* const* d_in, const int* in_sizes, int n_in,
                              void* d_out, int out_size, void* d_ws, size_t ws_size,
                              hipStream_t stream) {
  const float* x    = (const float*)d_in[0];
  const int*   ei   = (const int*)d_in[1];
  const float* Wl1  = (const float*)d_in[2];
  const float* bl1  = (const float*)d_in[3];
  const float* Wr1  = (const float*)d_in[4];
  const float* g1   = (const float*)d_in[5];
  const float* be1  = (const float*)d_in[6];
  const float* Wl2  = (const float*)d_in[7];
  const float* bl2  = (const float*)d_in[8];
  const float* Wr2  = (const float*)d_in[9];
  float* out = (float*)d_out;

  const int N = in_sizes[0] / DFEAT;  // 50000
  const int E = in_sizes[1] / 2;      // 800000
  const int* src = ei;
  const int* dst = ei + E;

  // Workspace layout
  float* agg      = (float*)d_ws;                  // [N,128]
  float* h        = agg + (size_t)N * DFEAT;       // [N,128] h_pre then h
  float* colsum   = h + (size_t)N * DFEAT;         // [128]
  float* colsumsq = colsum + DFEAT;                // [128]
  float* scale    = colsumsq + DFEAT;              // [128]
  float* shift    = scale + DFEAT;                 // [128]

  const long aggWords   = (long)N * DFEAT;
  const int  scatterBlk = (E * 32 + 255) / 256;
  const int  gemmBlocks = ((N / 16) + 3) / 4;      // 4 waves (tiles) per block

  // ---- Layer 1 ----
  fill_neginf_kernel<<<2048, 256, 0, stream>>>((unsigned int*)agg, aggWords);
  zero_kernel<<<1, 256, 0, stream>>>(colsum, 2 * DFEAT);
  scatter_max_kernel<<<scatterBlk, 256, 0, stream>>>(x, src, dst, agg, E);
  sage_gemm_wmma_kernel<128>
      <<<gemmBlocks, 128, (size_t)2 * 128 * LDSPAD * sizeof(float), stream>>>(
          agg, x, Wl1, Wr1, bl1, h, colsum, colsumsq, N);
  bn_finalize_kernel<<<1, 128, 0, stream>>>(colsum, colsumsq, g1, be1, scale, shift, N);
  bn_relu_kernel<<<(int)((aggWords / 4 + 255) / 256), 256, 0, stream>>>(h, scale, shift, aggWords);

  // ---- Layer 2 ----
  fill_neginf_kernel<<<2048, 256, 0, stream>>>((unsigned int*)agg, aggWords);
  scatter_max_kernel<<<scatterBlk, 256, 0, stream>>>(h, src, dst, agg, E);
  sage_gemm_wmma_kernel<32>
      <<<gemmBlocks, 128, (size_t)2 * 32 * LDSPAD * sizeof(float), stream>>>(
          agg, h, Wl2, Wr2, bl2, out, nullptr, nullptr, N);
}